// ODEBlock_33947421507707
// MI455X (gfx1250) — compile-verified
//
#include <hip/hip_runtime.h>

typedef __attribute__((ext_vector_type(2))) float v2f;
typedef __attribute__((ext_vector_type(4))) float v4f;
typedef __attribute__((ext_vector_type(8))) float v8f;

#define BATCH 8192
#define IN    784
#define INP   788   // padded LDS row stride (dwords): (l*788)%64 distinct for l=0..15
#define SYS   1024
#define NOUT  10
#define ROWS_PER_WG 16
#define THREADS 512

__device__ __forceinline__ v8f wmma_f32(v2f a, v2f b, v8f c) {
  // V_WMMA_F32_16X16X4_F32 : D = A(16x4 f32) * B(4x16 f32) + C(16x16 f32)
  return __builtin_amdgcn_wmma_f32_16x16x4_f32(
      /*neg_a=*/false, a, /*neg_b=*/false, b,
      /*c_mod=*/(short)0, c, /*reuse_a=*/false, /*reuse_b=*/false);
}

// One Lorenz-96 stage over this thread's 32 contiguous features (in regs).
// MODE 0: k1  (acc = k;       xt = x + cn*k)
// MODE 1: mid (acc += wgt*k;  xt = x + cn*k)
// MODE 2: k4  (x += h/6*(acc+k); xt = x)
template <int MODE>
__device__ __forceinline__ void lorenz_stage(float* xt, float* xs, float* acc,
                                             float wgt, float cn,
                                             float* halo, int row, int ln) {
  const int base = (row * 32 + ln) * 3;
  halo[base + 0] = xt[30];   // becomes right-neighbor's x[i-2]
  halo[base + 1] = xt[31];   // becomes right-neighbor's x[i-1]
  halo[base + 2] = xt[0];    // becomes left-neighbor's x[i+1]
  __syncthreads();
  const int pb = (row * 32 + ((ln + 31) & 31)) * 3;
  const int nb = (row * 32 + ((ln + 1) & 31)) * 3;
  float p2  = halo[pb + 0];  // x[s-2]
  float p1  = halo[pb + 1];  // x[s-1]
  float hp1 = halo[nb + 2];  // x[s+32]
  __syncthreads();
#pragma unroll
  for (int i = 0; i < 32; ++i) {
    float nxt = (i < 31) ? xt[i + 1] : hp1;
    float cur = xt[i];
    float k = (nxt - p2) * p1 - cur + 8.0f;  // (x[i+1]-x[i-2])*x[i-1] - x[i] + F
    p2 = p1; p1 = cur;
    if (MODE == 0)      acc[i]  = k;
    else if (MODE == 1) acc[i] += wgt * k;
    if (MODE < 2) {
      xt[i] = xs[i] + cn * k;
    } else {
      xs[i] += (0.01f / 6.0f) * (acc[i] + k);
      xt[i]  = xs[i];
    }
  }
}

__global__ __launch_bounds__(THREADS)
void odeblock_fused(const float* __restrict__ x,
                    const float* __restrict__ W1,
                    const float* __restrict__ b1,
                    const float* __restrict__ W2,
                    const float* __restrict__ b2,
                    float* __restrict__ out) {
  // 64 KB LDS, three lives: [x tile 16x788] -> [h tile 16x1024] -> [halo 16x32x3]
  __shared__ float smem[ROWS_PER_WG * SYS];

  const int tid = threadIdx.x;
  const int wv  = tid >> 5;   // wave id 0..15
  const int ln  = tid & 31;   // lane
  const int lh  = ln >> 4;    // lane half (K-pair select)
  const int ll  = ln & 15;    // lane low (M / N index)
  const int m0  = blockIdx.x * ROWS_PER_WG;

  // ---- Stage A (x tile) into LDS once: coalesced b128, one row per wave ----
  {
    const float* srcrow = x + (size_t)(m0 + wv) * IN;
    float*       dstrow = smem + wv * INP;
    for (int idx = ln; idx < IN / 4; idx += 32) {       // 196 float4 per row
      v4f v = *(const v4f*)(srcrow + idx * 4);
      *(v4f*)(dstrow + idx * 4) = v;
    }
  }
  __syncthreads();

  // ---------------- GEMM1: h = x @ W1^T + b1  (FP32 WMMA) ----------------
  v8f cacc[4];
#pragma unroll
  for (int t = 0; t < 4; ++t) cacc[t] = (v8f)(0.0f);

  const float* aLds  = smem + ll * INP;  // bank-conflict-free A fragments
  const int    nbase = wv * 64 + ll;     // wave's 64-column strip of W1
  for (int k0 = 0; k0 < IN; k0 += 16) {
    // prefetch next W1 chunk for each of the 4 column tiles (global_prefetch_b8)
#pragma unroll
    for (int t = 0; t < 4; ++t)
      __builtin_prefetch(W1 + (size_t)(nbase + t * 16) * IN + k0 + lh * 2 + 16, 0, 3);
#pragma unroll
    for (int k = k0; k < k0 + 16; k += 4) {
      const int kk = k + lh * 2;
      // A[16x4]: lanes0-15 K=k,k+1 ; lanes16-31 K=k+2,k+3 (ds_load_b64)
      v2f a = *(const v2f*)(aLds + kk);
#pragma unroll
      for (int t = 0; t < 4; ++t) {
        // B[4x16] = W1^T tile: B[k][n] = W1[n][k]  (aligned global b64)
        v2f b = *(const v2f*)(W1 + (size_t)(nbase + t * 16) * IN + kk);
        cacc[t] = wmma_f32(a, b, cacc[t]);
      }
    }
  }
  __syncthreads();   // all A reads done before smem is overwritten with h

  // C layout: VGPR r -> M = r + 8*lh, N = ll ; add bias, stage h into LDS
#pragma unroll
  for (int t = 0; t < 4; ++t) {
    const int   n    = wv * 64 + t * 16 + ll;
    const float bias = b1[n];
#pragma unroll
    for (int r = 0; r < 8; ++r) {
      const int m = r + lh * 8;
      smem[m * SYS + n] = cacc[t][r] + bias;
    }
  }
  __syncthreads();

  // ---------------- RK4 Lorenz-96, 100 steps, state in registers ----------
  const int row = wv;  // one wave == one batch row; lane == 32-feature chunk
  float xs[32], xt[32], acc[32];
#pragma unroll
  for (int i = 0; i < 32; ++i) {
    xs[i] = smem[row * SYS + ln * 32 + i];
    xt[i] = xs[i];
  }
  __syncthreads();           // everyone read h before smem reused as halo

  float* halo = smem;        // 16*32*3 floats = 6 KB of the 64 KB buffer
  for (int s = 0; s < 100; ++s) {
    lorenz_stage<0>(xt, xs, acc, 0.0f, 0.005f, halo, row, ln);  // k1
    lorenz_stage<1>(xt, xs, acc, 2.0f, 0.005f, halo, row, ln);  // k2
    lorenz_stage<1>(xt, xs, acc, 2.0f, 0.01f,  halo, row, ln);  // k3
    lorenz_stage<2>(xt, xs, acc, 0.0f, 0.0f,   halo, row, ln);  // k4 + combine
  }

  // ---------------- GEMM2 (wave-reduced dot) + log_softmax ----------------
  float logits[NOUT];
#pragma unroll
  for (int o = 0; o < NOUT; ++o) {
    float p = 0.0f;
    const float* w2r = W2 + (size_t)o * SYS + ln * 32;
#pragma unroll
    for (int i = 0; i < 32; ++i) p += xs[i] * w2r[i];
#pragma unroll
    for (int off = 16; off > 0; off >>= 1) p += __shfl_xor(p, off, 32);
    logits[o] = p + b2[o];
  }
  float mx = logits[0];
#pragma unroll
  for (int o = 1; o < NOUT; ++o) mx = fmaxf(mx, logits[o]);
  float se = 0.0f;
#pragma unroll
  for (int o = 0; o < NOUT; ++o) se += expf(logits[o] - mx);
  const float lse = mx + logf(se);
  if (ln < NOUT) out[(size_t)(m0 + row) * NOUT + ln] = logits[ln] - lse;
}

extern "C" void kernel_launch(void* const* d_in, const int* in_sizes, int n_in,
                              void* d_out, int out_size, void* d_ws, size_t ws_size,
                              hipStream_t stream) {
  const float* x  = (const float*)d_in[0];
  const float* W1 = (const float*)d_in[1];
  const float* b1 = (const float*)d_in[2];
  const float* W2 = (const float*)d_in[3];
  const float* b2 = (const float*)d_in[4];
  float* out = (float*)d_out;

  dim3 grid(BATCH / ROWS_PER_WG);   // 512 workgroups
  dim3 block(THREADS);              // 512 threads = 16 wave32
  hipLaunchKernelGGL(odeblock_fused, grid, block, 0, stream,
                     x, W1, b1, W2, b2, out);
}